// Decoder_35467839930861
// MI455X (gfx1250) — compile-verified
//
#include <hip/hip_runtime.h>
#include <cstdint>

// ---------------------------------------------------------------------------
// Problem constants (from reference)
// ---------------------------------------------------------------------------
#define NUM_AGENTS   1000000u
#define NUM_ABS      64u
#define PAIRS        (NUM_AGENTS / 2u)          // 500000 agent pairs (a, a+PAIRS)
#define HALF_ELEMS   (NUM_AGENTS * 32u)         // 32,000,000 (threefry half count)
#define ROWS_PER_BLK 64u                        // 64 threads = 2 waves (wave32)
#define ROW_PITCH    65u                        // 64 dwords + 1 pad dword (bank-conflict-free)

typedef __attribute__((ext_vector_type(4))) unsigned int u32x4;
typedef __attribute__((ext_vector_type(8))) int          i32x8;
typedef __attribute__((ext_vector_type(4))) int          i32x4;

// ---------------------------------------------------------------------------
// JAX-compatible threefry2x32 (20 rounds), usable on host (fold_in) and device
// ---------------------------------------------------------------------------
__host__ __device__ static inline uint32_t rotl32(uint32_t v, int r) {
    return (v << r) | (v >> (32 - r));
}

__host__ __device__ static inline void threefry2x32(uint32_t k0, uint32_t k1,
                                                    uint32_t x0, uint32_t x1,
                                                    uint32_t& o0, uint32_t& o1) {
    const uint32_t ks2 = k0 ^ k1 ^ 0x1BD11BDAu;
    x0 += k0; x1 += k1;
#define TFR(r) { x0 += x1; x1 = rotl32(x1, (r)); x1 ^= x0; }
    TFR(13) TFR(15) TFR(26) TFR(6)   x0 += k1;  x1 += ks2 + 1u;
    TFR(17) TFR(29) TFR(16) TFR(24)  x0 += ks2; x1 += k0  + 2u;
    TFR(13) TFR(15) TFR(26) TFR(6)   x0 += k0;  x1 += k1  + 3u;
    TFR(17) TFR(29) TFR(16) TFR(24)  x0 += k1;  x1 += ks2 + 4u;
    TFR(13) TFR(15) TFR(26) TFR(6)   x0 += ks2; x1 += k0  + 5u;
#undef TFR
    o0 = x0; o1 = x1;
}

// jax.random.uniform bit->float path: ((bits>>9)|0x3F800000) as f32 - 1, scaled
__device__ static inline float bits_to_uniform(uint32_t bits) {
    union { uint32_t u; float f; } cvt;
    cvt.u = (bits >> 9) | 0x3F800000u;
    const float mn = 1e-6f, mx = 1.0f - 1e-6f;
    float u = (cvt.f - 1.0f) * (mx - mn) + mn;
    return fmaxf(u, mn);
}

// ---------------------------------------------------------------------------
// Kernel: one thread per agent PAIR (a, a+PAIRS). Tiles of 64 logit rows for
// each half are DMA'd into LDS by the Tensor Data Mover with row padding.
// ---------------------------------------------------------------------------
__global__ __launch_bounds__(ROWS_PER_BLK)
void decoder_gumbel_kernel(const float* __restrict__ absActions,   // [64]
                           const float* __restrict__ logitsG,      // [1M,64]
                           const float* __restrict__ Wg,           // [2,2]
                           const float* __restrict__ Bg,           // [2]
                           float*       __restrict__ out,          // [3M]
                           uint32_t fk0, uint32_t fk1)             // folded key
{
    __shared__ float smA[ROWS_PER_BLK * ROW_PITCH];  // rows a        .. a+63
    __shared__ float smB[ROWS_PER_BLK * ROW_PITCH];  // rows a+PAIRS  ..
    __shared__ float sAbs[NUM_ABS];

    const uint32_t lt       = threadIdx.x;                  // 0..63
    const uint32_t rowStart = blockIdx.x * ROWS_PER_BLK;    // < PAIRS
    const uint32_t p        = rowStart + lt;                // pair index

    sAbs[lt] = absActions[lt];                              // 64 threads, 64 entries

#if defined(__gfx1250__) && __has_builtin(__builtin_amdgcn_tensor_load_to_lds)
    // Only wave 0 issues the TDM ops (TDM ignores EXEC -> must be a scalar branch).
    if (__builtin_amdgcn_readfirstlane(lt >> 5) == 0) {
        const uint32_t td1 = __builtin_amdgcn_readfirstlane(PAIRS - rowStart); // rows left (OOB rows -> 0)

        // ---- D# group1 (shared by both tiles) ----
        // data_size=4B(2), pad_enable=1, pad_interval=5 (every 64 dwords),
        // pad_amount=0 (1 dword) -> LDS row pitch 65 dwords, bank-conflict-free.
        i32x8 g1;
        g1[0] = (int)((2u << 16) | (1u << 20) | (5u << 22));
        g1[1] = (int)(64u << 16);                        // tensor_dim0[15:0] = 64
        g1[2] = (int)((td1 & 0xFFFFu) << 16);            // dim0 hi=0 | tensor_dim1 lo
        g1[3] = (int)((td1 >> 16) | (64u << 16));        // tensor_dim1 hi | tile_dim0=64
        g1[4] = (int)ROWS_PER_BLK;                       // tile_dim1 = 64 rows, tile_dim2 = 0
        g1[5] = 64;                                      // tensor_dim0_stride = 64 elems
        g1[6] = 0;
        g1[7] = 0;
        i32x4 g2 = {1, 1, 0, 0};                         // tensor_dim2=1, tensor_dim3=1
        i32x4 g3 = {0, (int)(1u << 16), 0, 0};           // tensor_dim4=1
        i32x8 gx = {0, 0, 0, 0, 0, 0, 0, 0};             // extra group (clang-23 6-arg form)

        const uint64_t gaA = (uint64_t)(uintptr_t)(logitsG + (uint64_t)rowStart * 64u);
        const uint64_t gaB = (uint64_t)(uintptr_t)(logitsG + ((uint64_t)rowStart + PAIRS) * 64u);
        const uint32_t ldsA = (uint32_t)(uintptr_t)(&smA[0]);   // low 32 bits = LDS offset
        const uint32_t ldsB = (uint32_t)(uintptr_t)(&smB[0]);

        u32x4 g0A = {1u, ldsA, (uint32_t)gaA, (uint32_t)(gaA >> 32) | (2u << 30)};
        u32x4 g0B = {1u, ldsB, (uint32_t)gaB, (uint32_t)(gaB >> 32) | (2u << 30)};

        __builtin_amdgcn_tensor_load_to_lds(g0A, g1, g2, g3, gx, 0);
        __builtin_amdgcn_tensor_load_to_lds(g0B, g1, g2, g3, gx, 0);
        __builtin_amdgcn_s_wait_tensorcnt(0);
    }
#else
    // Fallback (non-TDM toolchain / host parse): cooperative coalesced loads.
    {
        const uint32_t remain = PAIRS - rowStart;
        for (uint32_t r = 0; r < ROWS_PER_BLK; ++r) {
            float a = 0.0f, b = 0.0f;
            if (r < remain) {
                a = logitsG[((uint64_t)(rowStart + r)) * 64u + lt];
                b = logitsG[((uint64_t)(rowStart + r) + PAIRS) * 64u + lt];
            }
            smA[r * ROW_PITCH + lt] = a;
            smB[r * ROW_PITCH + lt] = b;
        }
    }
#endif
    __syncthreads();

    if (p < PAIRS) {
        const float* rowA = &smA[lt * ROW_PITCH];
        const float* rowB = &smB[lt * ROW_PITCH];
        const uint32_t baseCtr = p * 64u;                // < 2^25, exact

        float best0 = -3.402823466e38f, best1 = -3.402823466e38f;
        uint32_t idx0 = 0u, idx1 = 0u;

#pragma unroll 4
        for (uint32_t k = 0; k < 64u; ++k) {
            // One threefry call feeds BOTH agents of the pair (out0 / out1).
            uint32_t o0, o1;
            threefry2x32(fk0, fk1, baseCtr + k, baseCtr + k + HALF_ELEMS, o0, o1);
            const float gum0 = -__logf(-__logf(bits_to_uniform(o0)));
            const float gum1 = -__logf(-__logf(bits_to_uniform(o1)));
            const float s0 = rowA[k] + gum0;
            const float s1 = rowB[k] + gum1;
            if (s0 > best0) { best0 = s0; idx0 = k; }    // strict > == argmax first-max
            if (s1 > best1) { best1 = s1; idx1 = k; }
        }

        const float w00 = Wg[0], w01 = Wg[1], w10 = Wg[2], w11 = Wg[3];
        const float b0 = Bg[0], b1 = Bg[1];

        const uint32_t a0 = p, a1 = p + PAIRS;
        const float ga0 = sAbs[idx0], ga1 = sAbs[idx1];

        // agent a0: linear(2->2) + softmax + argmax
        {
            const float f  = (float)a0;
            const float l0 = w00 * ga0 + w01 * f + b0;
            const float l1 = w10 * ga0 + w11 * f + b1;
            const float m  = fmaxf(l0, l1);
            const float e0 = __expf(l0 - m), e1 = __expf(l1 - m);
            const float inv = 1.0f / (e0 + e1);
            ((float2*)out)[a0] = make_float2(e0 * inv, e1 * inv);
            out[2u * NUM_AGENTS + a0] = (l1 > l0) ? 1.0f : 0.0f;
        }
        // agent a1
        {
            const float f  = (float)a1;
            const float l0 = w00 * ga1 + w01 * f + b0;
            const float l1 = w10 * ga1 + w11 * f + b1;
            const float m  = fmaxf(l0, l1);
            const float e0 = __expf(l0 - m), e1 = __expf(l1 - m);
            const float inv = 1.0f / (e0 + e1);
            ((float2*)out)[a1] = make_float2(e0 * inv, e1 * inv);
            out[2u * NUM_AGENTS + a1] = (l1 > l0) ? 1.0f : 0.0f;
        }
    }
}

// ---------------------------------------------------------------------------
// Host launcher
// ---------------------------------------------------------------------------
extern "C" void kernel_launch(void* const* d_in, const int* in_sizes, int n_in,
                              void* d_out, int out_size, void* d_ws, size_t ws_size,
                              hipStream_t stream) {
    const float* absActions = (const float*)d_in[0];  // [64]
    const float* logitsG    = (const float*)d_in[1];  // [1M, 64]
    const float* Wg         = (const float*)d_in[2];  // [2, 2]
    const float* Bg         = (const float*)d_in[3];  // [2]
    float*       out        = (float*)d_out;          // [2M probs | 1M actions]

    // jax.random.fold_in(key(42), 7): key=(0,42), counter=(0,7) -> folded key.
    uint32_t fk0, fk1;
    threefry2x32(0u, 42u, 0u, 7u, fk0, fk1);

    const uint32_t nBlocks = (PAIRS + ROWS_PER_BLK - 1u) / ROWS_PER_BLK; // 7813
    decoder_gumbel_kernel<<<dim3(nBlocks), dim3(ROWS_PER_BLK), 0, stream>>>(
        absActions, logitsG, Wg, Bg, out, fk0, fk1);
}